// Cheb_GCNN_11785390260543
// MI455X (gfx1250) — compile-verified
//
#include <hip/hip_runtime.h>

#define NN   20000
#define EE   640000
#define IN_F 128
#define C1_F 256
#define C2_F 128
#define OUT_F 10

typedef __attribute__((ext_vector_type(2))) float v2f;
typedef __attribute__((ext_vector_type(8))) float v8f;

// ---------------- utility kernels ----------------

__global__ void zero_k(float* __restrict__ p, int n) {
    int i = blockIdx.x * blockDim.x + threadIdx.x;
    if (i < n) p[i] = 0.0f;
}

__global__ void negcopy_k(float* __restrict__ dst, const float* __restrict__ src, int n) {
    int i = blockIdx.x * blockDim.x + threadIdx.x;
    if (i < n) dst[i] = -src[i];
}

// ---------------- graph normalization ----------------

__global__ void edge_deg_k(const int* __restrict__ row, const int* __restrict__ col,
                           const float* __restrict__ ew, float* __restrict__ deg) {
    int e = blockIdx.x * blockDim.x + threadIdx.x;
    if (e >= EE) return;
    int r = row[e], c = col[e];
    if ((unsigned)r >= NN || (unsigned)c >= NN) return;
    if (r != c) atomicAdd(deg + r, ew[e]);
}

__global__ void deg_to_dis_k(float* __restrict__ deg) {
    int i = blockIdx.x * blockDim.x + threadIdx.x;
    if (i >= NN) return;
    float d = deg[i];
    deg[i] = (d > 0.0f) ? rsqrtf(d) : 0.0f;
}

__global__ void edge_norm_k(const int* __restrict__ row, const int* __restrict__ col,
                            const float* __restrict__ ew, const float* __restrict__ dis,
                            float* __restrict__ wn) {
    int e = blockIdx.x * blockDim.x + threadIdx.x;
    if (e >= EE) return;
    int r = row[e], c = col[e];
    if ((unsigned)r >= NN || (unsigned)c >= NN) { wn[e] = 0.0f; return; }
    // lambda_max = 2 -> factor (2/lambda_max) = 1, diagonal term = 0
    wn[e] = (r == c) ? 0.0f : -(dis[r] * ew[e] * dis[c]);
}

// ---------------- SpMM scatter: out[col] += scale*wn[e] * in[row] ----------------
// blockDim = 256; each thread handles one (edge, 4-feature chunk).
__global__ void scatter_k(const int* __restrict__ row, const int* __restrict__ col,
                          const float* __restrict__ wn, float scale,
                          const float* __restrict__ in, float* __restrict__ out,
                          int F, int lgF4) {
    int t = threadIdx.x;
    int F4m = (1 << lgF4) - 1;
    int e = (blockIdx.x << (8 - lgF4)) + (t >> lgF4);
    if (e >= EE) return;
    int r = row[e], c = col[e];
    if ((unsigned)r >= NN || (unsigned)c >= NN) return;
    float w = wn[e] * scale;
    int f = (t & F4m) << 2;
    float4 v = *(const float4*)(in + (size_t)r * F + f);   // b128 load
    float* op = out + (size_t)c * F + f;
    atomicAdd(op + 0, w * v.x);
    atomicAdd(op + 1, w * v.y);
    atomicAdd(op + 2, w * v.z);
    atomicAdd(op + 3, w * v.w);
}

// ---------------- weight repack into WMMA B-fragment order ----------------
// P[kc][tN][k4][lane] = ( W[kc][k4*4+kb][n], W[kc][k4*4+kb+1][n] )
//   with n = tN*16 + (lane&15), kb = 2*(lane>>4)
__global__ void packW_k(const float* __restrict__ W, float2* __restrict__ P,
                        int Fin, int Fout) {
    int idx = blockIdx.x * blockDim.x + threadIdx.x;
    int k4n = Fin >> 2;
    int tNn = Fout >> 4;
    int total = 4 * tNn * k4n * 32;
    if (idx >= total) return;
    int lane = idx & 31;
    int rest = idx >> 5;
    int k4 = rest % k4n;  rest /= k4n;
    int tN = rest % tNn;
    int kc = rest / tNn;
    int n = tN * 16 + (lane & 15);
    int k = (k4 << 2) + ((lane >> 4) << 1);
    const float* Wk = W + ((size_t)kc * Fin + k) * Fout + n;
    P[idx] = make_float2(Wk[0], Wk[Fout]);
}

// ---------------- WMMA f32 16x16x4 multi-GEMM + bias + ReLU ----------------
// Each wave computes a 16x64 output slab (4 accumulators), sharing A loads.
__device__ __forceinline__ void wmma_pass4(v8f acc[4], const float* __restrict__ A,
                                           const float2* __restrict__ B0,
                                           int k4n, int lane, int tileStride) {
    bool hi = (lane & 16) != 0;
    #pragma unroll 2
    for (int k4 = 0; k4 < k4n; ++k4) {
        float4 av = *(const float4*)(A + (k4 << 2));       // b128: A[m][k..k+3]
        v2f a;
        a.x = hi ? av.z : av.x;
        a.y = hi ? av.w : av.y;
        const float2* Bl = B0 + (k4 << 5) + lane;
        #pragma unroll
        for (int j = 0; j < 4; ++j) {
            float2 bv = Bl[(size_t)j * tileStride];        // coalesced b64
            v2f b; b.x = bv.x; b.y = bv.y;
            acc[j] = __builtin_amdgcn_wmma_f32_16x16x4_f32(false, a, false, b,
                                                           (short)0, acc[j],
                                                           false, false);
        }
    }
}

__global__ void cheb_gemm4_k(const float* __restrict__ T0, const float* __restrict__ T1,
                             const float* __restrict__ T2, const float* __restrict__ T3,
                             const float2* __restrict__ P,   // packed [4][Fout/16][Fin/4][32]
                             const float* __restrict__ bias,
                             float* __restrict__ out,
                             int Fin, int Fout, int totalWaves) {
    int lane = threadIdx.x & 31;
    int wv = blockIdx.x * (blockDim.x >> 5) + (threadIdx.x >> 5);
    if (wv >= totalWaves) return;          // wave-uniform: EXEC stays all-1s

    int nGroups = Fout >> 6;               // 64-column groups
    int tM = wv / nGroups;
    int ng = wv - tM * nGroups;

    int m = tM * 16 + (lane & 15);
    int k4n = Fin >> 2;
    int tileStride = k4n << 5;             // float2 elements per 16-col tile
    const float2* Pg = P + (size_t)(ng * 4) * tileStride;
    size_t PcStride = (size_t)(Fout >> 4) * tileStride;

    v8f acc[4] = {};
    wmma_pass4(acc, T0 + (size_t)m * Fin, Pg + 0 * PcStride, k4n, lane, tileStride);
    wmma_pass4(acc, T1 + (size_t)m * Fin, Pg + 1 * PcStride, k4n, lane, tileStride);
    wmma_pass4(acc, T2 + (size_t)m * Fin, Pg + 2 * PcStride, k4n, lane, tileStride);
    wmma_pass4(acc, T3 + (size_t)m * Fin, Pg + 3 * PcStride, k4n, lane, tileStride);

    // D layout: VGPR r -> row (r + 8*(lane>=16)), col = lane&15
    int mBase = tM * 16 + ((lane >> 4) << 3);
    #pragma unroll
    for (int j = 0; j < 4; ++j) {
        int n = (ng * 4 + j) * 16 + (lane & 15);
        float bv = bias[n];
        #pragma unroll
        for (int r = 0; r < 8; ++r) {
            float v = acc[j][r] + bv;
            v = v > 0.0f ? v : 0.0f;        // fused ReLU (applied before BN)
            out[(size_t)(mBase + r) * Fout + n] = v;
        }
    }
}

// ---------------- BatchNorm (training mode, population var) ----------------

__global__ void bn_stats_k(const float* __restrict__ x, const float* __restrict__ gamma,
                           const float* __restrict__ beta, float* __restrict__ scale,
                           float* __restrict__ shift, int F) {
    int f = blockIdx.x;
    float s = 0.0f, ss = 0.0f;
    for (int i = threadIdx.x; i < NN; i += blockDim.x) {
        float v = x[(size_t)i * F + f];
        s += v; ss += v * v;
    }
    __shared__ float sh1[256];
    __shared__ float sh2[256];
    sh1[threadIdx.x] = s; sh2[threadIdx.x] = ss;
    __syncthreads();
    for (int o = blockDim.x >> 1; o > 0; o >>= 1) {
        if (threadIdx.x < (unsigned)o) {
            sh1[threadIdx.x] += sh1[threadIdx.x + o];
            sh2[threadIdx.x] += sh2[threadIdx.x + o];
        }
        __syncthreads();
    }
    if (threadIdx.x == 0) {
        float mean = sh1[0] / (float)NN;
        float var  = sh2[0] / (float)NN - mean * mean;
        float sc   = gamma[f] * rsqrtf(var + 1e-5f);
        scale[f] = sc;
        shift[f] = beta[f] - mean * sc;
    }
}

__global__ void bn_apply_k(float* __restrict__ x, const float* __restrict__ scale,
                           const float* __restrict__ shift, int n, int Fmask) {
    int i = blockIdx.x * blockDim.x + threadIdx.x;
    if (i >= n) return;
    int f = i & Fmask;   // F is a power of two
    x[i] = x[i] * scale[f] + shift[f];
}

// ---------------- final linear head: out = H2 @ Wlin^T + blin ----------------

__global__ void linear_k(const float* __restrict__ h, const float* __restrict__ Wl,
                         const float* __restrict__ bl, float* __restrict__ out) {
    int nrow = blockIdx.x * blockDim.x + threadIdx.x;
    int o = blockIdx.y;
    if (nrow >= NN) return;
    const float* hp = h + (size_t)nrow * C2_F;
    const float* wp = Wl + (size_t)o * C2_F;
    float acc = 0.0f;
    #pragma unroll 8
    for (int i = 0; i < C2_F; ++i) acc += hp[i] * wp[i];
    out[(size_t)nrow * OUT_F + o] = acc + bl[o];
}

// ---------------- launch ----------------

extern "C" void kernel_launch(void* const* d_in, const int* in_sizes, int n_in,
                              void* d_out, int out_size, void* d_ws, size_t ws_size,
                              hipStream_t stream) {
    (void)in_sizes; (void)n_in; (void)out_size; (void)ws_size;
    const float* x   = (const float*)d_in[0];
    const int*   ei  = (const int*)d_in[1];
    const float* ew  = (const float*)d_in[2];
    const float* W1  = (const float*)d_in[3];
    const float* b1  = (const float*)d_in[4];
    const float* W2  = (const float*)d_in[5];
    const float* b2  = (const float*)d_in[6];
    const float* g1  = (const float*)d_in[7];
    const float* be1 = (const float*)d_in[8];
    const float* g2  = (const float*)d_in[9];
    const float* be2 = (const float*)d_in[10];
    const float* Wl  = (const float*)d_in[11];
    const float* bl  = (const float*)d_in[12];
    float* out = (float*)d_out;

    const int* row = ei;
    const int* col = ei + EE;

    float* ws  = (float*)d_ws;
    float* wn  = ws;                       // E
    float* deg = wn + EE;                  // N  (reused as dis)
    float* sc  = deg + NN;                 // 256
    float* sh  = sc + 256;                 // 256
    float* TA  = sh + 256;                 // N*256 (layer1 uses N*128 of it)
    float* TB  = TA + (size_t)NN * C1_F;   // N*256
    float* TC  = TB + (size_t)NN * C1_F;   // N*256
    float* H1  = TC + (size_t)NN * C1_F;   // N*256
    float* H2  = H1 + (size_t)NN * C1_F;   // N*128
    float2* P1 = (float2*)(H2 + (size_t)NN * C2_F);            // 4*128*256 floats
    float2* P2 = P1 + (size_t)2 * IN_F * C1_F;                 // 4*256*128 floats

    const int B = 256;
    // ---- graph normalization ----
    zero_k<<<(NN + B - 1) / B, B, 0, stream>>>(deg, NN);
    edge_deg_k<<<(EE + B - 1) / B, B, 0, stream>>>(row, col, ew, deg);
    deg_to_dis_k<<<(NN + B - 1) / B, B, 0, stream>>>(deg);
    edge_norm_k<<<(EE + B - 1) / B, B, 0, stream>>>(row, col, ew, deg, wn);

    // ---- weight repack (tiny) ----
    packW_k<<<(4 * IN_F * C1_F / 2 + B - 1) / B, B, 0, stream>>>(W1, P1, IN_F, C1_F);
    packW_k<<<(4 * C1_F * C2_F / 2 + B - 1) / B, B, 0, stream>>>(W2, P2, C1_F, C2_F);

    // ---- layer 1: Chebyshev recursion on F=128 (lgF4 = 5) ----
    {
        const int F = IN_F, lg = 5;
        int sgrid = (EE << lg) >> 8;  // EE*(F/4)/256 blocks, exact
        zero_k<<<(NN * F) / B, B, 0, stream>>>(TA, NN * F);
        scatter_k<<<sgrid, B, 0, stream>>>(row, col, wn, 1.0f, x, TA, F, lg);   // T1 = S x
        negcopy_k<<<(NN * F) / B, B, 0, stream>>>(TB, x, NN * F);
        scatter_k<<<sgrid, B, 0, stream>>>(row, col, wn, 2.0f, TA, TB, F, lg);  // T2 = 2 S T1 - x
        negcopy_k<<<(NN * F) / B, B, 0, stream>>>(TC, TA, NN * F);
        scatter_k<<<sgrid, B, 0, stream>>>(row, col, wn, 2.0f, TB, TC, F, lg);  // T3 = 2 S T2 - T1

        int waves = (NN / 16) * (C1_F / 64);    // 5000
        cheb_gemm4_k<<<(waves + 7) / 8, B, 0, stream>>>(x, TA, TB, TC, P1, b1, H1,
                                                        IN_F, C1_F, waves);
        bn_stats_k<<<C1_F, B, 0, stream>>>(H1, g1, be1, sc, sh, C1_F);
        bn_apply_k<<<(NN * C1_F) / B, B, 0, stream>>>(H1, sc, sh, NN * C1_F, C1_F - 1);
    }

    // ---- layer 2: Chebyshev recursion on F=256 (lgF4 = 6) ----
    {
        const int F = C1_F, lg = 6;
        int sgrid = (EE << lg) >> 8;
        zero_k<<<(NN * F) / B, B, 0, stream>>>(TA, NN * F);
        scatter_k<<<sgrid, B, 0, stream>>>(row, col, wn, 1.0f, H1, TA, F, lg);
        negcopy_k<<<(NN * F) / B, B, 0, stream>>>(TB, H1, NN * F);
        scatter_k<<<sgrid, B, 0, stream>>>(row, col, wn, 2.0f, TA, TB, F, lg);
        negcopy_k<<<(NN * F) / B, B, 0, stream>>>(TC, TA, NN * F);
        scatter_k<<<sgrid, B, 0, stream>>>(row, col, wn, 2.0f, TB, TC, F, lg);

        int waves = (NN / 16) * (C2_F / 64);    // 2500
        cheb_gemm4_k<<<(waves + 7) / 8, B, 0, stream>>>(H1, TA, TB, TC, P2, b2, H2,
                                                        C1_F, C2_F, waves);
        bn_stats_k<<<C2_F, B, 0, stream>>>(H2, g2, be2, sc, sh, C2_F);
        bn_apply_k<<<(NN * C2_F) / B, B, 0, stream>>>(H2, sc, sh, NN * C2_F, C2_F - 1);
    }

    // ---- output head ----
    dim3 lgrid((NN + B - 1) / B, OUT_F);
    linear_k<<<lgrid, B, 0, stream>>>(H2, Wl, bl, out);
}